// VQGAN_45758581572166
// MI455X (gfx1250) — compile-verified
//
#include <hip/hip_runtime.h>

typedef float v2f __attribute__((ext_vector_type(2)));
typedef float v8f __attribute__((ext_vector_type(8)));

#define NPIX 65536          // 16 * 64 * 64
#define CD   256
#define NK   1024
#define DEC_ELEMS 3145728   // 16*3*256*256

// ---------------------------------------------------------------------------
// Encoder: stride-4 4x4 conv, NCHW -> row-major [NPIX, 256] (NHWC flattened)
// one block per output pixel; 48-float patch in LDS; thread t = out channel
// ---------------------------------------------------------------------------
__global__ __launch_bounds__(256) void enc_kernel(const float* __restrict__ X,
                                                  const float* __restrict__ W,
                                                  const float* __restrict__ bvec,
                                                  float* __restrict__ ENC) {
    __shared__ float patch[48];
    int n = blockIdx.x;
    int b = n >> 12, rem = n & 4095, y = rem >> 6, xq = rem & 63;
    int t = threadIdx.x;
    if (t < 48) {
        int ci = t >> 4, q = t & 15, dy = q >> 2, dx = q & 3;
        patch[t] = X[(((size_t)(b * 3 + ci)) << 16) +
                     (size_t)(4 * y + dy) * 256 + (4 * xq + dx)];
    }
    __syncthreads();
    const float* wr = W + t * 48;
    float s = bvec[t];
#pragma unroll
    for (int j = 0; j < 48; j++) s += patch[j] * wr[j];
    ENC[((size_t)n << 8) + t] = s;
}

// ---------------------------------------------------------------------------
// codebook squared norms: one wave per code
// ---------------------------------------------------------------------------
__global__ __launch_bounds__(256) void cnorm_kernel(const float* __restrict__ CB,
                                                    float* __restrict__ CN) {
    int lane = threadIdx.x & 31, w = threadIdx.x >> 5;
    int k = blockIdx.x * 8 + w;
    const float* cr = CB + ((size_t)k << 8);
    float s = 0.f;
    for (int c = lane; c < 256; c += 32) { float v = cr[c]; s += v * v; }
    for (int m = 16; m >= 1; m >>= 1) s += __shfl_xor(s, m, 32);
    if (lane == 0) CN[k] = s;
}

// ---------------------------------------------------------------------------
// f32 WMMA GEMM:  Out = A(gathered) x B^T(+bias)   [MODE 0, 1x1 convs]
//                 or decoder scatter epilogue       [MODE 1]
// Each wave owns a 16 x (NT*16) output tile, K=256 in steps of 4.
// MODE 0: B(k,col) = Bp[col*256 + k]  (weights are [out,in] row-major)
// MODE 1: B(k,col) = Bp[k*48  + col]  (dec_w is [256, 3,4,4] = [256,48])
// ---------------------------------------------------------------------------
template <int NT, int MODE>
__global__ __launch_bounds__(128) void gemm_kernel(const float* __restrict__ A,
                                                   const int* __restrict__ gather,
                                                   const float* __restrict__ Bp,
                                                   const float* __restrict__ bias,
                                                   float* __restrict__ Out,
                                                   int ncb) {
    int lane = threadIdx.x & 31, w = threadIdx.x >> 5;
    int gw = blockIdx.x * 4 + w;
    int mtile = gw / ncb, cb = gw % ncb;
    int mbase = mtile * 16, nbase = cb * NT * 16;
    int half = lane >> 4, l16 = lane & 15;

    int arow = mbase + l16;
    if (gather) arow = gather[arow];
    const float* ap = A + ((size_t)arow << 8) + 2 * half;

    v8f acc[NT] = {};
    for (int kk = 0; kk < 256; kk += 4) {
        v2f a; a.x = ap[kk]; a.y = ap[kk + 1];
#pragma unroll
        for (int t = 0; t < NT; t++) {
            int col = nbase + t * 16 + l16;
            v2f bb;
            if (MODE == 0) {
                const float* bp = Bp + ((size_t)col << 8) + 2 * half;
                bb.x = bp[kk]; bb.y = bp[kk + 1];
            } else {
                int k0 = kk + 2 * half;
                bb.x = Bp[k0 * 48 + col];
                bb.y = Bp[(k0 + 1) * 48 + col];
            }
            acc[t] = __builtin_amdgcn_wmma_f32_16x16x4_f32(
                false, a, false, bb, (short)0, acc[t], false, false);
        }
    }

    int mr = mbase + 8 * half;  // this lane's first output row
    if (MODE == 0) {
#pragma unroll
        for (int t = 0; t < NT; t++) {
            int col = nbase + t * 16 + l16;
            float bc = bias[col];
#pragma unroll
            for (int v = 0; v < 8; v++)
                Out[((size_t)(mr + v) << 8) + col] = acc[t][v] + bc;
        }
    } else {
        // conv_transpose, stride==kernel==4, VALID: out[4y+ry,4x+rx] uses
        // weight (3-ry, 3-rx)  -> flip at scatter
#pragma unroll
        for (int t = 0; t < NT; t++) {
            int j = t * 16 + l16;            // 0..47 = o*16 + ky*4 + kx
            int o = j >> 4, q = j & 15, ky = q >> 2, kx = q & 3;
            int ry = 3 - ky, rx = 3 - kx;
            float bo = bias[o];
#pragma unroll
            for (int v = 0; v < 8; v++) {
                int row = mr + v;
                int b = row >> 12, rem = row & 4095, y = rem >> 6, xq = rem & 63;
                Out[(((size_t)(b * 3 + o)) << 16) +
                    (size_t)(4 * y + ry) * 256 + (4 * xq + rx)] = acc[t][v] + bo;
            }
        }
    }
}

// ---------------------------------------------------------------------------
// VQ: per 16-row tile, scores S = Z_tile x CB^T over all 1024 codes via WMMA,
// d = |c|^2 - 2*s ; argmin with first-index tie-break (matches jnp.argmin)
// 8 waves/block, wave w owns codebook columns [w*128, w*128+128)
// ---------------------------------------------------------------------------
__global__ __launch_bounds__(256) void vq_kernel(const float* __restrict__ Z,
                                                 const float* __restrict__ CB,
                                                 const float* __restrict__ CN,
                                                 int* __restrict__ idx_out) {
    __shared__ float lv[8][16];
    __shared__ int   li[8][16];
    int lane = threadIdx.x & 31, w = threadIdx.x >> 5;
    int half = lane >> 4, l16 = lane & 15;
    int mbase = blockIdx.x * 16;
    int nbase = w * 128;

    const float* ap = Z + ((size_t)(mbase + l16) << 8) + 2 * half;

    v8f acc[8] = {};
    for (int kk = 0; kk < 256; kk += 4) {
        v2f a; a.x = ap[kk]; a.y = ap[kk + 1];
#pragma unroll
        for (int t = 0; t < 8; t++) {
            int col = nbase + t * 16 + l16;
            const float* bp = CB + ((size_t)col << 8) + 2 * half;
            v2f bb; bb.x = bp[kk]; bb.y = bp[kk + 1];
            acc[t] = __builtin_amdgcn_wmma_f32_16x16x4_f32(
                false, a, false, bb, (short)0, acc[t], false, false);
        }
    }

    float cn[8];
#pragma unroll
    for (int t = 0; t < 8; t++) cn[t] = CN[nbase + t * 16 + l16];

    float rb[8]; int ri[8];
#pragma unroll
    for (int v = 0; v < 8; v++) {
        float best = 3.402823466e38f; int bi = 0;
#pragma unroll
        for (int t = 0; t < 8; t++) {
            float d = cn[t] - 2.0f * acc[t][v];
            int ci = nbase + t * 16 + l16;
            if (d < best) { best = d; bi = ci; }  // ascending ci: '<' keeps first
        }
        // butterfly across the 16-lane group sharing this M row-set
        for (int m = 1; m <= 8; m <<= 1) {
            float ov = __shfl_xor(best, m, 32);
            int   oi = __shfl_xor(bi,  m, 32);
            if (ov < best || (ov == best && oi < bi)) { best = ov; bi = oi; }
        }
        rb[v] = best; ri[v] = bi;
    }
    if (l16 == 0) {
#pragma unroll
        for (int v = 0; v < 8; v++) {
            lv[w][8 * half + v] = rb[v];
            li[w][8 * half + v] = ri[v];
        }
    }
    __syncthreads();
    if (threadIdx.x < 16) {
        int m = threadIdx.x;
        float best = lv[0][m]; int bi = li[0][m];
        for (int ww = 1; ww < 8; ww++) {         // ascending col blocks -> first-min
            float ov = lv[ww][m]; int oi = li[ww][m];
            if (ov < best || (ov == best && oi < bi)) { best = ov; bi = oi; }
        }
        idx_out[mbase + m] = bi;
    }
}

// ---------------------------------------------------------------------------
// loss partials: SSE of (z - codebook[idx]) per block, deterministic order
// ---------------------------------------------------------------------------
__global__ __launch_bounds__(256) void loss_kernel(const float* __restrict__ Z,
                                                   const float* __restrict__ CB,
                                                   const int* __restrict__ idx,
                                                   float* __restrict__ part) {
    __shared__ float wsum[8];
    int lane = threadIdx.x & 31, w = threadIdx.x >> 5;
    int gw = blockIdx.x * 8 + w;
    int tw = gridDim.x * 8;
    float s = 0.f;
    for (int r = gw; r < NPIX; r += tw) {
        const float* zr = Z + ((size_t)r << 8);
        const float* cr = CB + ((size_t)idx[r] << 8);
        float ls = 0.f;
        for (int c = lane; c < 256; c += 32) {
            float d = zr[c] - cr[c];
            ls += d * d;
        }
        s += ls;
    }
    for (int m = 16; m >= 1; m >>= 1) s += __shfl_xor(s, m, 32);
    if (lane == 0) wsum[w] = s;
    __syncthreads();
    if (threadIdx.x == 0) {
        float t = 0.f;
        for (int i = 0; i < 8; i++) t += wsum[i];
        part[blockIdx.x] = t;
    }
}

__global__ void loss_final(const float* __restrict__ part, float* __restrict__ out) {
    int lane = threadIdx.x;
    float s = 0.f;
    for (int i = 0; i < 32; i++) s += part[lane + 32 * i];
    for (int m = 16; m >= 1; m >>= 1) s += __shfl_xor(s, m, 32);
    if (lane == 0) out[0] = s * (1.25f / (65536.0f * 256.0f));
}

// ---------------------------------------------------------------------------
extern "C" void kernel_launch(void* const* d_in, const int* in_sizes, int n_in,
                              void* d_out, int out_size, void* d_ws, size_t ws_size,
                              hipStream_t stream) {
    const float* x     = (const float*)d_in[0];
    const float* enc_w = (const float*)d_in[1];
    const float* enc_b = (const float*)d_in[2];
    const float* w1    = (const float*)d_in[3];
    const float* b1    = (const float*)d_in[4];
    const float* cbk   = (const float*)d_in[5];
    const float* w2    = (const float*)d_in[6];
    const float* b2    = (const float*)d_in[7];
    const float* dw    = (const float*)d_in[8];
    const float* db    = (const float*)d_in[9];

    float* ENC  = (float*)d_ws;                       // [NPIX,256]; reused as dec_in
    float* Zb   = ENC + (size_t)NPIX * CD;            // [NPIX,256]
    float* CN   = Zb + (size_t)NPIX * CD;             // [1024]
    float* PART = CN + NK;                            // [1024]

    float* dec_out  = (float*)d_out;                  // [16,3,256,256]
    int*   idx_out  = (int*)d_out + DEC_ELEMS;        // [16,64,64] as int32 bits
    float* loss_out = (float*)d_out + DEC_ELEMS + NPIX;

    enc_kernel<<<NPIX, 256, 0, stream>>>(x, enc_w, enc_b, ENC);
    gemm_kernel<4, 0><<<4096, 128, 0, stream>>>(ENC, nullptr, w1, b1, Zb, 4);
    cnorm_kernel<<<128, 256, 0, stream>>>(cbk, CN);
    vq_kernel<<<4096, 256, 0, stream>>>(Zb, cbk, CN, idx_out);
    gemm_kernel<4, 0><<<4096, 128, 0, stream>>>(cbk, idx_out, w2, b2, ENC, 4);
    gemm_kernel<3, 1><<<1024, 128, 0, stream>>>(ENC, nullptr, dw, db, dec_out, 1);
    loss_kernel<<<1024, 256, 0, stream>>>(Zb, cbk, idx_out, PART);
    loss_final<<<1, 32, 0, stream>>>(PART, loss_out);
}